// SSGConv_9466107920387
// MI455X (gfx1250) — compile-verified
//
#include <hip/hip_runtime.h>
#include <math.h>

#define SSGC_ALPHA 0.1f
#define SSGC_K 5

typedef __attribute__((ext_vector_type(2))) float v2f;
typedef __attribute__((ext_vector_type(8))) float v8f;

// ---------------- degree / normalization ----------------

__global__ void deg_init_kernel(float* __restrict__ deg, int N) {
    int i = blockIdx.x * blockDim.x + threadIdx.x;
    if (i < N) deg[i] = 1.0f;  // self-loop weight 1 pre-added
}

__global__ void deg_scatter_kernel(const long long* __restrict__ col,
                                   const float* __restrict__ w,
                                   float* __restrict__ deg, int E) {
    int e = blockIdx.x * blockDim.x + threadIdx.x;
    if (e < E) atomicAdd(&deg[(int)col[e]], w[e]);
}

__global__ void dinv_kernel(float* __restrict__ deg, int N) {
    int i = blockIdx.x * blockDim.x + threadIdx.x;
    if (i < N) {
        float d = deg[i];
        deg[i] = (d > 0.0f) ? rsqrtf(d) : 0.0f;  // in place: deg -> dinv
    }
}

__global__ void edge_norm_kernel(const long long* __restrict__ row,
                                 const long long* __restrict__ col,
                                 const float* __restrict__ w,
                                 const float* __restrict__ dinv,
                                 float* __restrict__ wn, int E) {
    int e = blockIdx.x * blockDim.x + threadIdx.x;
    if (e < E) wn[e] = dinv[(int)row[e]] * w[e] * dinv[(int)col[e]];
}

// ---------------- SSGC propagation ----------------

__global__ void xacc_init_kernel(const float4* __restrict__ x,
                                 float4* __restrict__ xacc, int n4) {
    int i = blockIdx.x * blockDim.x + threadIdx.x;
    if (i < n4) {
        float4 v = x[i];
        float4 o;
        o.x = SSGC_ALPHA * v.x; o.y = SSGC_ALPHA * v.y;
        o.z = SSGC_ALPHA * v.z; o.w = SSGC_ALPHA * v.w;
        xacc[i] = o;
    }
}

// hout[n][:] = dinv[n]^2 * hin[n][:]   (self-loop term; also initializes hout)
__global__ void selfloop_init_kernel(const float4* __restrict__ hin,
                                     float4* __restrict__ hout,
                                     const float* __restrict__ dinv, int n4) {
    int i = blockIdx.x * blockDim.x + threadIdx.x;  // i indexes float4 of [N,64]
    if (i < n4) {
        int n = i >> 4;  // 16 float4 per node row
        float dv = dinv[n];
        float s = dv * dv;
        float4 v = hin[i];
        float4 o;
        o.x = s * v.x; o.y = s * v.y; o.z = s * v.z; o.w = s * v.w;
        hout[i] = o;
    }
}

// 16 consecutive threads per edge, each owns 4 floats -> 256B coalesced gather
__global__ void spmm_scatter_kernel(const long long* __restrict__ row,
                                    const long long* __restrict__ col,
                                    const float* __restrict__ wn,
                                    const float* __restrict__ h,
                                    float* __restrict__ hnew, int E) {
    long long t = (long long)blockIdx.x * blockDim.x + threadIdx.x;
    int e = (int)(t >> 4);
    if (e >= E) return;
    int d = ((int)t & 15) * 4;
    int rs = (int)row[e];
    int cs = (int)col[e];
    float w = wn[e];
    float4 hv = *(const float4*)(h + (long long)rs * 64 + d);
    float* dst = hnew + (long long)cs * 64 + d;
    atomicAdd(dst + 0, w * hv.x);
    atomicAdd(dst + 1, w * hv.y);
    atomicAdd(dst + 2, w * hv.z);
    atomicAdd(dst + 3, w * hv.w);
}

__global__ void accum_kernel(const float4* __restrict__ h,
                             float4* __restrict__ xacc, float coef, int n4) {
    int i = blockIdx.x * blockDim.x + threadIdx.x;
    if (i < n4) {
        float4 v = h[i];
        float4 a = xacc[i];
        a.x += coef * v.x; a.y += coef * v.y;
        a.z += coef * v.z; a.w += coef * v.w;
        xacc[i] = a;
    }
}

// ---------------- final GEMM: out[N,64] = xacc[N,64] @ W[64,64] + b ----------------
// One wave computes one 16x16 tile of the output using V_WMMA_F32_16X16X4_F32,
// 16 chained WMMAs over K=64. 4 waves per block cover the 4 column tiles.
__global__ __launch_bounds__(128) void gemm_wmma_kernel(const float* __restrict__ X,
                                                        const float* __restrict__ W,
                                                        const float* __restrict__ bias,
                                                        float* __restrict__ out, int N) {
    int wave = threadIdx.x >> 5;   // 0..3 -> column tile
    int lane = threadIdx.x & 31;
    int m0 = blockIdx.x * 16;
    if (m0 >= N) return;           // uniform per block
    int n0 = wave * 16;
    int r = lane & 15;
    int half = lane >> 4;          // 0 or 1

    int mrow = m0 + r;
    if (mrow >= N) mrow = N - 1;   // clamp loads; stores guarded below
    const float* xrow = X + (long long)mrow * 64;

    v8f c = {};
    #pragma unroll
    for (int k0 = 0; k0 < 64; k0 += 4) {
        int ka = k0 + half * 2;
        v2f a, b;
        // A 16x4 f32: VGPR0 = K {0 | 2}, VGPR1 = K {1 | 3} per half-lane group
        a[0] = xrow[ka + 0];
        a[1] = xrow[ka + 1];
        // B 4x16 f32: row-striped across lanes; VGPR0 = K {0 | 2}, VGPR1 = K {1 | 3}
        b[0] = W[(ka + 0) * 64 + n0 + r];
        b[1] = W[(ka + 1) * 64 + n0 + r];
        c = __builtin_amdgcn_wmma_f32_16x16x4_f32(
                /*neg_a=*/false, a, /*neg_b=*/false, b,
                /*c_mod=*/(short)0, c, /*reuse_a=*/false, /*reuse_b=*/false);
    }

    float bv = bias[n0 + r];
    #pragma unroll
    for (int j = 0; j < 8; ++j) {
        int m = m0 + half * 8 + j;   // C/D: VGPR j holds M = j (+8 for upper lanes)
        if (m < N) out[(long long)m * 64 + n0 + r] = c[j] + bv;
    }
}

// ---------------- launch ----------------

extern "C" void kernel_launch(void* const* d_in, const int* in_sizes, int n_in,
                              void* d_out, int out_size, void* d_ws, size_t ws_size,
                              hipStream_t stream) {
    const float*     x    = (const float*)d_in[0];       // [N,64]
    const long long* eidx = (const long long*)d_in[1];   // [2,E] int64
    const float*     ew   = (const float*)d_in[2];       // [E]
    const float*     W    = (const float*)d_in[3];       // [64,64]
    const float*     bias = (const float*)d_in[4];       // [64]

    const int N = in_sizes[0] / 64;
    const int E = in_sizes[2];
    const long long* row = eidx;       // source
    const long long* col = eidx + E;   // target

    // workspace layout (floats): dinv[N] | wn[E] | h0[N*64] | h1[N*64] | xacc[N*64]
    float* dinv = (float*)d_ws;
    float* wn   = dinv + N;
    float* h0   = wn + E;
    float* h1   = h0 + (size_t)N * 64;
    float* xacc = h1 + (size_t)N * 64;

    const int B = 256;
    const int n4 = N * 16;  // float4 elements of an [N,64] array

    deg_init_kernel   <<<(N + B - 1) / B, B, 0, stream>>>(dinv, N);
    deg_scatter_kernel<<<(E + B - 1) / B, B, 0, stream>>>(col, ew, dinv, E);
    dinv_kernel       <<<(N + B - 1) / B, B, 0, stream>>>(dinv, N);
    edge_norm_kernel  <<<(E + B - 1) / B, B, 0, stream>>>(row, col, ew, dinv, wn, E);

    xacc_init_kernel  <<<(n4 + B - 1) / B, B, 0, stream>>>((const float4*)x, (float4*)xacc, n4);

    const float coef = (1.0f - SSGC_ALPHA) / (float)SSGC_K;
    long long tscatter = (long long)E * 16;
    int scat_blocks = (int)((tscatter + B - 1) / B);

    const float* hin = x;
    float* hout = h0;
    float* other = h1;
    for (int k = 0; k < SSGC_K; ++k) {
        selfloop_init_kernel<<<(n4 + B - 1) / B, B, 0, stream>>>(
            (const float4*)hin, (float4*)hout, dinv, n4);
        spmm_scatter_kernel<<<scat_blocks, B, 0, stream>>>(row, col, wn, hin, hout, E);
        accum_kernel<<<(n4 + B - 1) / B, B, 0, stream>>>(
            (const float4*)hout, (float4*)xacc, coef, n4);
        hin = hout;
        float* t = hout; hout = other; other = t;
    }

    gemm_wmma_kernel<<<(N + 15) / 16, 128, 0, stream>>>(xacc, W, bias, (float*)d_out, N);
}